// GCN_61409442398709
// MI455X (gfx1250) — compile-verified
//
#include <hip/hip_runtime.h>
#include <hip/hip_bf16.h>
#include <math.h>

// ---------------------------------------------------------------------------
// GCN forward on gfx1250 (CDNA5). fp32 WMMA (V_WMMA_F32_16X16X4_F32) for both
// dense layers (register-blocked 16x64 per wave in layer 1); float-atomic
// scatter for the normalized message passing (L2-resident accumulators).
// ---------------------------------------------------------------------------

typedef __attribute__((ext_vector_type(2))) float v2f;
typedef __attribute__((ext_vector_type(8))) float v8f;

#define N_FEAT   128
#define HIDDEN   128
#define N_CLS    10
#define N_CLSP   16   // padded class dim for the 16x16 WMMA tile

// ---------------- degree / normalization ----------------------------------

__global__ void gcn_init_deg(float* deg, int n) {
    int i = blockIdx.x * blockDim.x + threadIdx.x;
    if (i < n) deg[i] = 1.0f;   // self-loop contributes 1 to every node
}

__global__ void gcn_count_deg(const int* __restrict__ col, float* deg, int e) {
    int i = blockIdx.x * blockDim.x + threadIdx.x;
    if (i < e) atomicAdd(&deg[col[i]], 1.0f);
}

__global__ void gcn_rsqrt_deg(float* deg, int n) {
    int i = blockIdx.x * blockDim.x + threadIdx.x;
    if (i < n) deg[i] = rsqrtf(deg[i]);   // deg >= 1 always
}

// ---------------- GEMM1: H1 = X @ W1 + b1  (N x 128) -----------------------
// One wave computes a 16(M) x 64(N) register block: one A-frag feeds 4 WMMAs
// per k-step. Two waves per block cover the full 128-wide row.

__global__ void gcn_gemm1_wmma(const float* __restrict__ X,
                               const float* __restrict__ W,
                               const float* __restrict__ bias,
                               float* __restrict__ H, int n) {
    const int wave = threadIdx.x >> 5;   // 0..1
    const int lane = threadIdx.x & 31;
    const int half = lane >> 4;          // 0: K=k..k+1, 1: K=k+2..k+3
    const int l16  = lane & 15;
    const int m0   = blockIdx.x * 16;
    const int n0   = wave * 64;

    int mr = m0 + l16; if (mr > n - 1) mr = n - 1;   // clamped A row (reads only)
    const float* __restrict__ xrow = X + (size_t)mr * N_FEAT;

    v8f acc0 = {}, acc1 = {}, acc2 = {}, acc3 = {};
    #pragma unroll 2
    for (int k = 0; k < N_FEAT; k += 4) {
        const int k0 = k + 2 * half;
        v2f a; a.x = xrow[k0]; a.y = xrow[k0 + 1];
        const float* bp = W + (size_t)k0 * HIDDEN + n0 + l16;
        v2f b0; b0.x = bp[0];  b0.y = bp[HIDDEN];
        v2f b1; b1.x = bp[16]; b1.y = bp[HIDDEN + 16];
        v2f b2; b2.x = bp[32]; b2.y = bp[HIDDEN + 32];
        v2f b3; b3.x = bp[48]; b3.y = bp[HIDDEN + 48];
        acc0 = __builtin_amdgcn_wmma_f32_16x16x4_f32(false, a, false, b0, (short)0, acc0, false, false);
        acc1 = __builtin_amdgcn_wmma_f32_16x16x4_f32(false, a, false, b1, (short)0, acc1, false, false);
        acc2 = __builtin_amdgcn_wmma_f32_16x16x4_f32(false, a, false, b2, (short)0, acc2, false, false);
        acc3 = __builtin_amdgcn_wmma_f32_16x16x4_f32(false, a, false, b3, (short)0, acc3, false, false);
    }

    const float bv0 = bias[n0 + l16];
    const float bv1 = bias[n0 + 16 + l16];
    const float bv2 = bias[n0 + 32 + l16];
    const float bv3 = bias[n0 + 48 + l16];
    const int mrow0 = m0 + half * 8;

    if (m0 + 16 <= n) {                      // wave-uniform fast path
        #pragma unroll
        for (int v = 0; v < 8; ++v) {
            float* hp = H + (size_t)(mrow0 + v) * HIDDEN + n0 + l16;
            hp[0]  = acc0[v] + bv0;
            hp[16] = acc1[v] + bv1;
            hp[32] = acc2[v] + bv2;
            hp[48] = acc3[v] + bv3;
        }
    } else {                                 // cold tail (never taken for N%16==0)
        for (int v = 0; v < 8; ++v) {
            int m = mrow0 + v;
            if (m < n) {
                float* hp = H + (size_t)m * HIDDEN + n0 + l16;
                hp[0]  = acc0[v] + bv0;
                hp[16] = acc1[v] + bv1;
                hp[32] = acc2[v] + bv2;
                hp[48] = acc3[v] + bv3;
            }
        }
    }
}

// ---------------- scatter layer-1: out1[c] += H1[r] * dis[r]*dis[c] --------
// One wave per (edge or self-loop); each lane handles 4 consecutive features.

__global__ void gcn_scatter1(const float* __restrict__ H,
                             const int* __restrict__ row,
                             const int* __restrict__ col,
                             const float* __restrict__ dis,
                             float* __restrict__ out, int e, int n) {
    long long gid = (long long)blockIdx.x * blockDim.x + threadIdx.x;
    long long edge = gid >> 5;
    int lane = (int)(gid & 31);
    if (edge >= (long long)e + n) return;
    int r, c;
    if (edge < e) { r = row[edge]; c = col[edge]; }
    else          { r = c = (int)(edge - e); }
    const float nrm = dis[r] * dis[c];
    const float4 hv = *((const float4*)(H + (size_t)r * HIDDEN) + lane);
    float* op = out + (size_t)c * HIDDEN + lane * 4;
    atomicAdd(op + 0, hv.x * nrm);
    atomicAdd(op + 1, hv.y * nrm);
    atomicAdd(op + 2, hv.z * nrm);
    atomicAdd(op + 3, hv.w * nrm);
}

// ---------------- GEMM2: H2 = out1 @ W2 + b2  (N x 16 padded) --------------
// Single N-tile (10 real cols); 4 waves/block cover 4 M-tiles.

__global__ void gcn_gemm2_wmma(const float* __restrict__ A,
                               const float* __restrict__ W2,
                               const float* __restrict__ b2,
                               float* __restrict__ H2, int n, int mtiles) {
    const int wave = threadIdx.x >> 5;
    const int lane = threadIdx.x & 31;
    const int half = lane >> 4;
    const int l16  = lane & 15;
    const int mt   = blockIdx.x * 4 + wave;
    if (mt >= mtiles) return;            // wave-uniform: EXEC stays all-1s
    const int m0 = mt * 16;

    int mr = m0 + l16; if (mr > n - 1) mr = n - 1;
    const bool nv = (l16 < N_CLS);
    const float* __restrict__ arow = A + (size_t)mr * HIDDEN;

    v8f acc = {};
    #pragma unroll 4
    for (int k = 0; k < HIDDEN; k += 4) {
        const int k0 = k + 2 * half;
        v2f a; a.x = arow[k0]; a.y = arow[k0 + 1];
        v2f b;
        b.x = nv ? W2[(size_t)k0 * N_CLS + l16] : 0.0f;
        b.y = nv ? W2[(size_t)(k0 + 1) * N_CLS + l16] : 0.0f;
        acc = __builtin_amdgcn_wmma_f32_16x16x4_f32(
            false, a, false, b, (short)0, acc, false, false);
    }

    const float bv = nv ? b2[l16] : 0.0f;
    const int mrow0 = m0 + half * 8;

    if (m0 + 16 <= n) {                      // wave-uniform fast path
        #pragma unroll
        for (int v = 0; v < 8; ++v)
            H2[(size_t)(mrow0 + v) * N_CLSP + l16] = acc[v] + bv;
    } else {
        for (int v = 0; v < 8; ++v) {
            int m = mrow0 + v;
            if (m < n) H2[(size_t)m * N_CLSP + l16] = acc[v] + bv;
        }
    }
}

// ---------------- scatter layer-2: out2[c] += H2[r] * norm (10 feats) ------

__global__ void gcn_scatter2(const float* __restrict__ H2,
                             const int* __restrict__ row,
                             const int* __restrict__ col,
                             const float* __restrict__ dis,
                             float* __restrict__ out2, int e, int n) {
    long long gid = (long long)blockIdx.x * blockDim.x + threadIdx.x;
    long long edge = gid >> 4;
    int lane = (int)(gid & 15);
    if (edge >= (long long)e + n || lane >= N_CLS) return;
    int r, c;
    if (edge < e) { r = row[edge]; c = col[edge]; }
    else          { r = c = (int)(edge - e); }
    const float nrm = dis[r] * dis[c];
    atomicAdd(out2 + (size_t)c * N_CLSP + lane,
              H2[(size_t)r * N_CLSP + lane] * nrm);
}

// ---------------- pooled sums + counts -------------------------------------

__global__ void gcn_pool(const float* __restrict__ out2,
                         const int* __restrict__ batch,
                         float* __restrict__ pooled,
                         float* __restrict__ cnt, int n) {
    long long gid = (long long)blockIdx.x * blockDim.x + threadIdx.x;
    long long node = gid >> 4;
    int lane = (int)(gid & 15);
    if (node >= n) return;
    int g = batch[node];
    if (lane < N_CLS)
        atomicAdd(&pooled[(size_t)g * N_CLS + lane],
                  out2[(size_t)node * N_CLSP + lane]);
    else if (lane == 15)
        atomicAdd(&cnt[g], 1.0f);
}

// ---------------- mean + log_softmax ---------------------------------------

__global__ void gcn_finalize(const float* __restrict__ pooled,
                             const float* __restrict__ cnt,
                             float* __restrict__ out, int G) {
    int g = blockIdx.x * blockDim.x + threadIdx.x;
    if (g >= G) return;
    float c = fmaxf(cnt[g], 1.0f);
    float v[N_CLS];
    float m = -INFINITY;
    #pragma unroll
    for (int f = 0; f < N_CLS; ++f) {
        v[f] = pooled[(size_t)g * N_CLS + f] / c;
        m = fmaxf(m, v[f]);
    }
    float s = 0.0f;
    #pragma unroll
    for (int f = 0; f < N_CLS; ++f) s += expf(v[f] - m);
    float lse = logf(s);
    #pragma unroll
    for (int f = 0; f < N_CLS; ++f) out[(size_t)g * N_CLS + f] = v[f] - m - lse;
}

// ---------------------------------------------------------------------------

extern "C" void kernel_launch(void* const* d_in, const int* in_sizes, int n_in,
                              void* d_out, int out_size, void* d_ws, size_t ws_size,
                              hipStream_t stream) {
    const float* x     = (const float*)d_in[0];
    const int*   ei    = (const int*)d_in[1];   // [2, E] flat: row then col
    const int*   batch = (const int*)d_in[2];
    // d_in[3] = num_graphs scalar (derived from out_size instead)
    const float* W1 = (const float*)d_in[4];
    const float* b1 = (const float*)d_in[5];
    const float* W2 = (const float*)d_in[6];
    const float* b2 = (const float*)d_in[7];
    float* out = (float*)d_out;

    const int N = in_sizes[0] / N_FEAT;   // 100000
    const int E = in_sizes[1] / 2;        // 1600000
    const int G = out_size / N_CLS;       // 64
    const int* row = ei;
    const int* col = ei + E;

    // Workspace layout (floats)
    float* h1     = (float*)d_ws;                      // N*128
    float* out1   = h1   + (size_t)N * HIDDEN;         // N*128
    float* h2     = out1 + (size_t)N * HIDDEN;         // N*16
    float* out2   = h2   + (size_t)N * N_CLSP;         // N*16
    float* dis    = out2 + (size_t)N * N_CLSP;         // N (deg -> rsqrt in place)
    float* pooled = dis  + (size_t)N;                  // G*10
    float* cnt    = pooled + (size_t)G * N_CLS;        // G (contiguous w/ pooled)

    // Zero the accumulators (graph-capturable async memsets)
    hipMemsetAsync(out1,   0, (size_t)N * HIDDEN * sizeof(float), stream);
    hipMemsetAsync(out2,   0, (size_t)N * N_CLSP * sizeof(float), stream);
    hipMemsetAsync(pooled, 0, (size_t)G * (N_CLS + 1) * sizeof(float), stream);

    const int mtiles = (N + 15) / 16;     // 6250

    // Degree / normalization
    gcn_init_deg <<<(N + 255) / 256, 256, 0, stream>>>(dis, N);
    gcn_count_deg<<<(E + 255) / 256, 256, 0, stream>>>(col, dis, E);
    gcn_rsqrt_deg<<<(N + 255) / 256, 256, 0, stream>>>(dis, N);

    // Layer 1: GEMM (16x64 register block per wave, 2 waves/block) + scatter
    gcn_gemm1_wmma<<<mtiles, 64, 0, stream>>>(x, W1, b1, h1, N);
    {
        long long t = (long long)(E + N) * 32;
        gcn_scatter1<<<(unsigned)((t + 255) / 256), 256, 0, stream>>>(
            h1, row, col, dis, out1, E, N);
    }

    // Layer 2
    gcn_gemm2_wmma<<<(mtiles + 3) / 4, 128, 0, stream>>>(out1, W2, b2, h2, N, mtiles);
    {
        long long t = (long long)(E + N) * 16;
        gcn_scatter2<<<(unsigned)((t + 255) / 256), 256, 0, stream>>>(
            h2, row, col, dis, out2, E, N);
    }

    // Pool + log_softmax
    {
        long long t = (long long)N * 16;
        gcn_pool<<<(unsigned)((t + 255) / 256), 256, 0, stream>>>(
            out2, batch, pooled, cnt, N);
    }
    gcn_finalize<<<(G + 63) / 64, 64, 0, stream>>>(pooled, cnt, out, G);
}